// MessagePassing_88974542503970
// MI455X (gfx1250) — compile-verified
//
#include <hip/hip_runtime.h>

typedef float v2f __attribute__((ext_vector_type(2)));
typedef float v8f __attribute__((ext_vector_type(8)));

#define FEAT 64

// ---------------------------------------------------------------------------
// Zero-init the output (harness poisons d_out with 0xAA before timing).
// ---------------------------------------------------------------------------
__global__ void __launch_bounds__(256) zero_f4_kernel(float4* __restrict__ out,
                                                      int n4) {
  int i = blockIdx.x * blockDim.x + threadIdx.x;
  const int stride = gridDim.x * blockDim.x;
  float4 z;
  z.x = z.y = z.z = z.w = 0.0f;
  for (; i < n4; i += stride) out[i] = z;
}

// ---------------------------------------------------------------------------
// Main scatter kernel. One wave32 handles 16 edges per batch iteration:
//   M[16,64] = diag(w[16]) @ X_gathered[16,64]    (v_wmma_f32_16x16x4_f32)
//   out[tgt[m], :] += M[m, :]                      (global_atomic_add_f32)
//
// WMMA fragment layouts (wave32, 32-bit data, CDNA5 ISA 7.12.2):
//   A (16x4):  lane -> M = lane&15 ; A.x -> K = 2*(lane>>4), A.y -> K+1
//   B (4x16):  lane -> N = lane&15 ; B.x -> K = 2*(lane>>4), B.y -> K+1
//   C/D(16x16):lane -> N = lane&15 ; vgpr r -> M = r + 8*(lane>>4)
//
// All addresses are carried as 32-bit byte offsets (x and out are each only
// 12.8 MB) so VMEM ops use the GVS saddr mode (SGPR base + 32-bit VGPR
// offset) and the per-tile +t*64B folds into the 24-bit IOFFSET.
// ---------------------------------------------------------------------------
__global__ void __launch_bounds__(256) mp_scatter_wmma_kernel(
    const float* __restrict__ x, const long long* __restrict__ src,
    const long long* __restrict__ tgt, const float* __restrict__ w,
    float* __restrict__ out, int nbatch) {
  const int lane = threadIdx.x & 31;
  const int half = lane >> 4;  // 0: lanes 0-15, 1: lanes 16-31
  const int m16 = lane & 15;
  const int col4 = m16 * 4;  // byte offset of this lane's column
  const int gwave = (blockIdx.x * blockDim.x + threadIdx.x) >> 5;
  const int nwave = (gridDim.x * blockDim.x) >> 5;

  const char* xb = (const char*)x;
  char* ob = (char*)out;

  for (int b = gwave; b < nbatch; b += nwave) {
    const int e0 = b << 4;

    // Prefetch next batch's edge metadata (lowers to global_prefetch_b8).
    if (b + nwave < nbatch) {
      const int en = (b + nwave) << 4;
      __builtin_prefetch(src + en, 0, 0);
      __builtin_prefetch(tgt + en, 0, 0);
      __builtin_prefetch(w + en, 0, 0);
    }

    // Edge metadata for this 16-edge batch (lanes 16-31 mirror 0-15; these
    // 128B chunks are broadcast hits in L0/L2). Convert node ids to row
    // byte offsets (row = 64 floats = 256 B) BEFORE shuffling.
    const int s256 = ((int)src[e0 + m16]) << 8;
    const int t256 = ((int)tgt[e0 + m16]) << 8;
    const float w_m = w[e0 + m16];

    // A = diag(w) (16x16) split into four 16x4 K-slices in WMMA A layout.
    v2f A[4];
#pragma unroll
    for (int k = 0; k < 4; ++k) {
      const int kx = 4 * k + 2 * half;  // global K of A.x for this lane
      A[k].x = (m16 == kx) ? w_m : 0.0f;
      A[k].y = (m16 == kx + 1) ? w_m : 0.0f;
    }

    // Gather byte offsets for B fragments (K indexes the 16 source rows).
    unsigned gx[4], gy[4];
#pragma unroll
    for (int k = 0; k < 4; ++k) {
      const int kx = 4 * k + 2 * half;
      gx[k] = (unsigned)(__shfl(s256, kx, 32) + col4);
      gy[k] = (unsigned)(__shfl(s256, kx + 1, 32) + col4);
    }

    // Scatter byte offsets: vgpr r of the C/D tile is output row
    // m = r + 8*half, column m16.
    unsigned so[8];
#pragma unroll
    for (int r = 0; r < 8; ++r) {
      so[r] = (unsigned)(__shfl(t256, r + 8 * half, 32) + col4);
    }

    // Four 16-column output tiles; chain 4 WMMAs (K=16 total) per tile.
#pragma unroll
    for (int t = 0; t < 4; ++t) {
      v8f c = {};
#pragma unroll
      for (int k = 0; k < 4; ++k) {
        v2f bf;
        // Half-wave reads 64 contiguous bytes of a gathered row: coalesced.
        bf.x = *(const float*)(xb + gx[k] + t * 64);
        bf.y = *(const float*)(xb + gy[k] + t * 64);
        c = __builtin_amdgcn_wmma_f32_16x16x4_f32(false, A[k], false, bf,
                                                  (short)0, c, false, false);
      }
#pragma unroll
      for (int r = 0; r < 8; ++r) {
        __hip_atomic_fetch_add((float*)(ob + so[r] + t * 64), c[r],
                               __ATOMIC_RELAXED, __HIP_MEMORY_SCOPE_AGENT);
      }
    }
  }
}

// ---------------------------------------------------------------------------
// Tail kernel for E % 16 != 0 (not needed for E = 1,600,000, kept for safety).
// ---------------------------------------------------------------------------
__global__ void __launch_bounds__(256) mp_tail_kernel(
    const float* __restrict__ x, const long long* __restrict__ src,
    const long long* __restrict__ tgt, const float* __restrict__ w,
    float* __restrict__ out, int e_begin, int e_end) {
  int i = blockIdx.x * blockDim.x + threadIdx.x;
  const int total = (e_end - e_begin) * FEAT;
  const int stride = gridDim.x * blockDim.x;
  for (; i < total; i += stride) {
    const int e = e_begin + i / FEAT;
    const int f = i % FEAT;
    const int s = (int)src[e];
    const int t = (int)tgt[e];
    const float v = x[(size_t)s * FEAT + f] * w[e];
    __hip_atomic_fetch_add(out + (size_t)t * FEAT + f, v, __ATOMIC_RELAXED,
                           __HIP_MEMORY_SCOPE_AGENT);
  }
}

// ---------------------------------------------------------------------------
// Launch: inputs are x [N*F f32], edge_index [2*E i64], edge_weights [E f32].
// ---------------------------------------------------------------------------
extern "C" void kernel_launch(void* const* d_in, const int* in_sizes, int n_in,
                              void* d_out, int out_size, void* d_ws,
                              size_t ws_size, hipStream_t stream) {
  const float* x = (const float*)d_in[0];
  const long long* ei = (const long long*)d_in[1];
  const float* w = (const float*)d_in[2];
  float* out = (float*)d_out;

  const int E = in_sizes[1] / 2;
  const long long* src = ei;      // edge_index[0, :]
  const long long* tgt = ei + E;  // edge_index[1, :]

  // 1) zero-init output
  const int n4 = out_size / 4;
  int zb = (n4 + 255) / 256;
  if (zb > 4096) zb = 4096;
  zero_f4_kernel<<<zb, 256, 0, stream>>>((float4*)out, n4);

  // 2) main WMMA scatter-add
  const int nbatch = E / 16;
  if (nbatch > 0) {
    int blocks = (nbatch + 7) / 8;  // 8 waves per block, 1 batch per wave
    if (blocks > 8192) blocks = 8192;
    mp_scatter_wmma_kernel<<<blocks, 256, 0, stream>>>(x, src, tgt, w, out,
                                                       nbatch);
  }

  // 3) tail (E % 16 == 0 here, so this never launches)
  const int rem = E - nbatch * 16;
  if (rem > 0) {
    int tb = (rem * FEAT + 255) / 256;
    mp_tail_kernel<<<tb, 256, 0, stream>>>(x, src, tgt, w, out, nbatch * 16,
                                           E);
  }
}